// FastSpeech2Decoder_10144712753619
// MI455X (gfx1250) — compile-verified
//
#include <hip/hip_runtime.h>
#include <hip/hip_bf16.h>
#include <stdint.h>

#define B_ROWS   4096
#define T_FRAMES 8192
#define T_TOKENS 2048
#define THREADS  256
#define TOK_PER_THREAD (T_TOKENS / THREADS)   // 8

typedef unsigned int v4u __attribute__((ext_vector_type(4)));
typedef int          v8i __attribute__((ext_vector_type(8)));
typedef int          v4i __attribute__((ext_vector_type(4)));

__global__ __launch_bounds__(THREADS) void avg_scalar_by_duration(
    const float* __restrict__ frames,
    const int*   __restrict__ frame_len,
    const int*   __restrict__ duration,
    const int*   __restrict__ dur_len,
    float*       __restrict__ out_tok,
    int*         __restrict__ out_dlen)
{
  __shared__ float s_frames[T_FRAMES];          // 32 KB staged by TDM
  __shared__ int   s_wave[THREADS / 32];

  const int row  = blockIdx.x;
  const int tid  = threadIdx.x;
  const int lane = tid & 31;                    // wave32
  const int wid  = tid >> 5;

  // ---- 1. Issue TDM async copy: one frame row, global -> LDS (wave 0 only;
  //         tensor ops ignore EXEC, so issuing from all 8 waves would DMA 8x).
  if (wid == 0) {
    uint64_t ga  = (uint64_t)(uintptr_t)(frames + (size_t)row * T_FRAMES);
    uint32_t lds = (uint32_t)(uintptr_t)(void*)&s_frames[0];
    v4u g0;
    g0.x = 0x1u;                                        // count=1 valid, user mode
    g0.y = lds;                                         // lds_addr (bytes)
    g0.z = (uint32_t)ga;                                // global_addr[31:0]
    g0.w = (uint32_t)((ga >> 32) & 0x01FFFFFFu)         // global_addr[56:32]
         | (2u << 30);                                  // type = 2 ("image")
    v8i g1;
    g1[0] = (int)(2u << 16);                            // data_size = 4 B
    g1[1] = (int)((uint32_t)T_FRAMES << 16);            // tensor_dim0[15:0] = 8192
    g1[2] = (int)(1u << 16);                            // tensor_dim1 = 1
    g1[3] = (int)((uint32_t)T_FRAMES << 16);            // tile_dim0 = 8192
    g1[4] = 1;                                          // tile_dim1 = 1, tile_dim2 = 0
    g1[5] = T_FRAMES;                                   // tensor_dim0_stride
    g1[6] = (int)((uint32_t)T_FRAMES << 16);            // tensor_dim1_stride[15:0]
    g1[7] = 0;
    v4i gz4 = {0, 0, 0, 0};                             // groups 2/3 unused (1-D)
    v8i gz8 = {0, 0, 0, 0, 0, 0, 0, 0};                 // extra group (6-arg form)
    __builtin_amdgcn_tensor_load_to_lds(g0, g1, gz4, gz4, gz8, /*cpol=*/0);
  }

  // ---- 2. Load this thread's 8 contiguous durations (2x b128) + local scan.
  const int* drow = duration + (size_t)row * T_TOKENS + tid * TOK_PER_THREAD;
  int4 da = ((const int4*)drow)[0];
  int4 db = ((const int4*)drow)[1];
  int d[TOK_PER_THREAD] = {da.x, da.y, da.z, da.w, db.x, db.y, db.z, db.w};
  int c[TOK_PER_THREAD];
  int run = 0;
#pragma unroll
  for (int j = 0; j < TOK_PER_THREAD; ++j) {
    int dj = d[j] > 0 ? d[j] : 0;                       // jnp.maximum(duration, 0)
    run += dj;
    c[j] = run;                                         // local inclusive cumsum
  }
  const int total = run;

  // ---- 3. Exclusive scan over thread totals: wave32 shfl scan + wave totals.
  int incl = total;
#pragma unroll
  for (int off = 1; off < 32; off <<= 1) {
    int n = __shfl_up(incl, off, 32);
    if (lane >= off) incl += n;
  }
  if (lane == 31) s_wave[wid] = incl;
  __syncthreads();
  int excl = incl - total;                              // lower lanes in wave
#pragma unroll
  for (int w = 0; w < THREADS / 32; ++w)
    if (w < wid) excl += s_wave[w];                     // lower waves

  // ---- 4. Wait for the TDM tile, then publish to all waves.
  if (wid == 0) __builtin_amdgcn_s_wait_tensorcnt(0);
  __syncthreads();

  // ---- 5. Segment means straight from LDS (each segment <= 7 frames).
  const int L  = frame_len[row];
  const int DL = dur_len[row];
  float v[TOK_PER_THREAD];
  int gstart = excl;
#pragma unroll
  for (int j = 0; j < TOK_PER_THREAD; ++j) {
    int gend = excl + c[j];
    int s0   = gstart < L ? gstart : L;                 // clipped cumsum boundaries
    int s1   = gend  < L ? gend  : L;
    int len  = s1 - s0;
    int tok  = tid * TOK_PER_THREAD + j;
    float val = 0.0f;
    if (len > 0 && tok < DL) {
      float acc = 0.0f;
      for (int k = s0; k < s1; ++k) acc += s_frames[k]; // exact serial sum
      val = acc / (float)len;
    }
    v[j] = val;
    gstart = gend;
  }

  float* orow = out_tok + (size_t)row * T_TOKENS + tid * TOK_PER_THREAD;
  ((float4*)orow)[0] = make_float4(v[0], v[1], v[2], v[3]);
  ((float4*)orow)[1] = make_float4(v[4], v[5], v[6], v[7]);

  if (tid == 0) out_dlen[row] = DL;                     // passthrough output #2
}

extern "C" void kernel_launch(void* const* d_in, const int* in_sizes, int n_in,
                              void* d_out, int out_size, void* d_ws, size_t ws_size,
                              hipStream_t stream) {
  const float* frames    = (const float*)d_in[0];   // (4096, 8192) f32
  const int*   frame_len = (const int*)d_in[1];     // (4096,) i32
  const int*   duration  = (const int*)d_in[2];     // (4096, 2048) i32
  const int*   dur_len   = (const int*)d_in[3];     // (4096,) i32
  float* out_tok  = (float*)d_out;                                   // (4096, 2048)
  int*   out_dlen = (int*)(out_tok + (size_t)B_ROWS * T_TOKENS);     // (4096,) raw i32 bits

  avg_scalar_by_duration<<<B_ROWS, THREADS, 0, stream>>>(
      frames, frame_len, duration, dur_len, out_tok, out_dlen);
}